// RWKV_40106404610572
// MI455X (gfx1250) — compile-verified
//
#include <hip/hip_runtime.h>
#include <hip/hip_bf16.h>
#include <stdint.h>

typedef __attribute__((ext_vector_type(2))) float v2f;
typedef __attribute__((ext_vector_type(8))) float v8f;

static __device__ __forceinline__ v8f wmma_f32_16x16x4(v2f a, v2f b, v8f c) {
  // (neg_a, A, neg_b, B, c_mod, C, reuse_a, reuse_b)
  return __builtin_amdgcn_wmma_f32_16x16x4_f32(false, a, false, b, (short)0, c,
                                               false, false);
}

// Async DMA: memory -> LDS, tracked by ASYNCcnt (no VGPR round trip).
static __device__ __forceinline__ void async_load_b128(unsigned lds_off,
                                                       const void* gaddr) {
  asm volatile("global_load_async_to_lds_b128 %0, %1, off"
               :: "v"(lds_off), "v"(gaddr) : "memory");
}

#define B_ 2
#define H_ 16
#define T_ 2048
#define K_ 64
#define V_ 64
#define CH 16  // chunk length == WMMA tile M/N

__launch_bounds__(128, 1)
__global__ void rwkv_chunk_wmma(const float* __restrict__ gr,
                                const float* __restrict__ gk,
                                const float* __restrict__ gv,
                                const float* __restrict__ gw,
                                const float* __restrict__ gu,
                                const float* __restrict__ gs0,
                                float* __restrict__ gout,
                                float* __restrict__ gsout) {
  __shared__ __align__(16) float shS[64][65];        // running state K x V
  __shared__ __align__(16) float shBuf[2][4][CH][64]; // double-buffered r,k,v,w
  __shared__ __align__(16) float shA[CH][65];        // r_t * prod_{s<t} w
  __shared__ __align__(16) float shB[CH][65];        // k_t / prod_{s<=t} w
  __shared__ __align__(16) float shM[CH][17];        // masked intra matrix
  __shared__ float shWc[64];                         // chunk-final decay
  __shared__ float shU[64];
  __shared__ float shD[CH];                          // diagonal r*(u.k)

  const int tid  = threadIdx.x;
  const int lane = tid & 31;
  const int wv   = tid >> 5;       // wave -> owns V columns [16*wv, 16*wv+16)
  const int hi   = lane >> 4;      // lane half (ISA fragment layouts)
  const int col  = lane & 15;
  const int c0   = wv * 16;
  const int bh   = blockIdx.x;     // 0..31
  const int h    = bh & (H_ - 1);

  const float* rp = gr + (size_t)bh * T_ * K_;
  const float* kp = gk + (size_t)bh * T_ * K_;
  const float* vp = gv + (size_t)bh * T_ * V_;
  const float* wp = gw + (size_t)h * T_ * K_;       // w broadcast over batch

  if (tid < 64) shU[tid] = gu[h * K_ + tid];
  for (int i = tid; i < K_ * V_; i += 128)
    shS[i >> 6][i & 63] = gs0[(size_t)bh * K_ * V_ + i];

  // ---- prologue: async-stage chunk 0 into buffer 0 (8 loads per wave) ----
  {
    float (*buf)[CH][64] = shBuf[0];
    for (int i = tid; i < CH * 16; i += 128) {
      int tt = i >> 4;
      int cc = (i & 15) << 2;
      size_t go = (size_t)tt * 64 + cc;
      async_load_b128((unsigned)(size_t)&buf[0][tt][cc], rp + go);
      async_load_b128((unsigned)(size_t)&buf[1][tt][cc], kp + go);
      async_load_b128((unsigned)(size_t)&buf[2][tt][cc], vp + go);
      async_load_b128((unsigned)(size_t)&buf[3][tt][cc], wp + go);
    }
  }

  for (int t0 = 0; t0 < T_; t0 += CH) {
    const int p = (t0 >> 4) & 1;

    // ---- issue async stage of NEXT chunk into other buffer, then wait for
    //      current buffer (in-order async completion: <=8 outstanding) ----
    if (t0 + CH < T_) {
      float (*buf)[CH][64] = shBuf[p ^ 1];
      for (int i = tid; i < CH * 16; i += 128) {
        int tt = i >> 4;
        int cc = (i & 15) << 2;
        size_t go = (size_t)(t0 + CH + tt) * 64 + cc;
        async_load_b128((unsigned)(size_t)&buf[0][tt][cc], rp + go);
        async_load_b128((unsigned)(size_t)&buf[1][tt][cc], kp + go);
        async_load_b128((unsigned)(size_t)&buf[2][tt][cc], vp + go);
        async_load_b128((unsigned)(size_t)&buf[3][tt][cc], wp + go);
      }
      asm volatile("s_wait_asynccnt 0x8" ::: "memory");
    } else {
      asm volatile("s_wait_asynccnt 0x0" ::: "memory");
    }
    __syncthreads();

    const float (*shR)[64] = shBuf[p][0];
    const float (*shK)[64] = shBuf[p][1];
    const float (*shV)[64] = shBuf[p][2];
    const float (*shW)[64] = shBuf[p][3];

    // ---- decay prefix products + u-diagonal (scalar prologue) ----
    if (tid < 64) {  // one thread per k index
      float pr = 1.0f;
      #pragma unroll
      for (int t = 0; t < CH; ++t) {
        shA[t][tid] = shR[t][tid] * pr;  // r_t * P_{t-1}
        pr *= shW[t][tid];
        shB[t][tid] = shK[t][tid] / pr;  // k_t / P_t
      }
      shWc[tid] = pr;                    // P_{C-1}
    } else if (tid < 64 + CH) {
      int i = tid - 64;
      float d = 0.f;
      #pragma unroll 8
      for (int kk = 0; kk < 64; ++kk) d += shR[i][kk] * shU[kk] * shK[i][kk];
      shD[i] = d;
    }
    __syncthreads();

    // ---- hoist reusable fragments into registers ----
    v2f af[16];  // A rows (used by M loop and inter loop)
    #pragma unroll
    for (int kk = 0; kk < 16; ++kk) {
      int kb = kk * 4 + 2 * hi;
      af[kk][0] = shA[col][kb]; af[kk][1] = shA[col][kb + 1];
    }
    v2f vf[4];   // V columns c0..c0+15 (used by intra and 4 state tiles)
    #pragma unroll
    for (int kk = 0; kk < 4; ++kk) {
      int kb = kk * 4 + 2 * hi;
      vf[kk][0] = shV[kb][c0 + col]; vf[kk][1] = shV[kb + 1][c0 + col];
    }

    // ---- intra matrix M = A @ B^T  (16x16, contract K=64) ----
    v8f m = {};
    #pragma unroll
    for (int kk = 0; kk < 16; ++kk) {
      int kb = kk * 4 + 2 * hi;
      v2f b; b[0] = shB[col][kb]; b[1] = shB[col][kb + 1];  // B[K][n]=Bm[n][K]
      m = wmma_f32_16x16x4(af[kk], b, m);
    }
    // causal mask (strict lower) + u-diagonal; wave 0 publishes
    if (wv == 0) {
      #pragma unroll
      for (int g = 0; g < 8; ++g) {
        int row = g + 8 * hi;                   // C/D layout row
        shM[row][col] = (col < row) ? m[g] : ((col == row) ? shD[row] : 0.f);
      }
    }
    __syncthreads();

    // ---- output tile: o = M @ Vc  +  A @ S[:, c0:c0+16) ----
    v8f o = {};
    #pragma unroll
    for (int kk = 0; kk < 4; ++kk) {
      int kb = kk * 4 + 2 * hi;
      v2f a; a[0] = shM[col][kb]; a[1] = shM[col][kb + 1];
      o = wmma_f32_16x16x4(a, vf[kk], o);
    }
    #pragma unroll
    for (int kk = 0; kk < 16; ++kk) {
      int kb = kk * 4 + 2 * hi;
      v2f b; b[0] = shS[kb][c0 + col]; b[1] = shS[kb + 1][c0 + col];
      o = wmma_f32_16x16x4(af[kk], b, o);
    }

    // ---- state update: S = Wc ⊙ (S + B^T @ Vc), wave-private columns ----
    #pragma unroll
    for (int kt = 0; kt < 4; ++kt) {
      v8f s;
      #pragma unroll
      for (int g = 0; g < 8; ++g) s[g] = shS[kt * 16 + g + 8 * hi][c0 + col];
      #pragma unroll
      for (int kk = 0; kk < 4; ++kk) {
        int tb = kk * 4 + 2 * hi;                    // contraction dim = t
        v2f a; a[0] = shB[tb][kt * 16 + col]; a[1] = shB[tb + 1][kt * 16 + col];
        s = wmma_f32_16x16x4(a, vf[kk], s);
      }
      #pragma unroll
      for (int g = 0; g < 8; ++g) {
        int krow = kt * 16 + g + 8 * hi;
        shS[krow][c0 + col] = s[g] * shWc[krow];
      }
    }

    // ---- write output rows of this chunk ----
    float* op = gout + ((size_t)bh * T_ + t0) * V_;
    #pragma unroll
    for (int g = 0; g < 8; ++g) {
      int trow = g + 8 * hi;
      op[(size_t)trow * V_ + c0 + col] = o[g];
    }
    __syncthreads();  // chunk-complete before buffer p is re-staged
  }

  // ---- final state ----
  float* sp = gsout + (size_t)bh * K_ * V_;
  for (int i = tid; i < K_ * V_; i += 128)
    sp[i] = shS[i >> 6][i & 63];
}

extern "C" void kernel_launch(void* const* d_in, const int* in_sizes, int n_in,
                              void* d_out, int out_size, void* d_ws, size_t ws_size,
                              hipStream_t stream) {
  const float* r  = (const float*)d_in[0];
  const float* k  = (const float*)d_in[1];
  const float* v  = (const float*)d_in[2];
  const float* w  = (const float*)d_in[3];
  const float* u  = (const float*)d_in[4];
  const float* s0 = (const float*)d_in[5];
  float* out  = (float*)d_out;
  float* sout = out + (size_t)B_ * H_ * T_ * V_;
  rwkv_chunk_wmma<<<dim3(B_ * H_), dim3(128), 0, stream>>>(r, k, v, w, u, s0,
                                                           out, sout);
}